// MultiTaskGNN_2353642078677
// MI455X (gfx1250) — compile-verified
//
#include <hip/hip_runtime.h>
#include <hip/hip_bf16.h>

typedef __attribute__((ext_vector_type(2))) float v2f;
typedef __attribute__((ext_vector_type(8))) float v8f;

#define NGRAPHS 50
#define HDIM 128
#define BN_EPS 1e-5f

// ---------------------------------------------------------------- utilities
__global__ void fill_f32(float* __restrict__ p, float v, int n) {
    int i = blockIdx.x * blockDim.x + threadIdx.x;
    if (i < n) p[i] = v;
}

// ---------------------------------------------------------------- degrees
__global__ void deg_scatter(const int* __restrict__ dst, float* __restrict__ deg, int E) {
    int e = blockIdx.x * blockDim.x + threadIdx.x;
    if (e < E) atomicAdd(&deg[dst[e]], 1.0f);
}

__global__ void deg_finalize(float* __restrict__ deg, int N) {
    int i = blockIdx.x * blockDim.x + threadIdx.x;
    if (i < N) deg[i] = rsqrtf(fmaxf(deg[i], 1.0f));   // becomes dinv
}

// ---------------------------------------------------------------- WMMA GEMM
// H[N x 128] = A[N x K] * W[K x 128], K in {8, 128}.
// Block: 256 threads = 8 waves; block tile 32 rows x 64 cols
// (grid = (ceil(N/32), 2)); wave (rt, ct) owns the 16x16 tile at
// rows row0+16*rt, cols colBase+16*ct.
// A tile staged via async global->LDS DMA (GLOBAL_LOAD_ASYNC_TO_LDS_B128,
// ASYNCcnt) while the W slice is repacked through VGPRs into a paired-K LDS
// layout Wp[(k>>1)*128 + col*2 + (k&1)] so a B fragment (W[ka][col],
// W[ka+1][col]) is one conflict-free ds_load_b64. A tile uses padded stride
// (K+4) -> conflict-free b64 reads (16B-aligned rows: (K+4)*4 = 528 = 33*16).
// Fragment packing per CDNA5 ISA 32-bit layouts:
//   A 16x4 : lanes 0-15 -> M, v0/v1 = K0/K1 ; lanes 16-31 -> K2/K3
//   B 4x16 : v0 = rows K0|K2, v1 = rows K1|K3, N striped across lanes
//   C 16x16: v[j] = row (j + 8*kh), col = lane&15
__global__ __launch_bounds__(256) void gemm_wmma_f32(
        const float* __restrict__ A, const float* __restrict__ W,
        float* __restrict__ H, int N, int K) {
    __shared__ float As[32 * (HDIM + 4)];   // 32 x K tile, stride K+4
    __shared__ float Wp[64 * HDIM];         // (K/2) x 64 pairs, stride 128

    const int tid  = threadIdx.x;
    const int wave = tid >> 5;
    const int lane = tid & 31;
    const int r    = lane & 15;          // M (A) / N (B,C) index within tile
    const int kh   = lane >> 4;          // K-half selector
    const int rt   = wave >> 2;          // row tile 0..1
    const int ct   = wave & 3;           // col tile 0..3
    const int row0 = blockIdx.x * 32;
    const int colBase = blockIdx.y * 64;
    const int strideA = K + 4;

    // ---- stage A tile: async global->LDS b128 per lane (no VGPR round-trip)
    const int quadsA = (32 * K) >> 2;
    const int kq = K >> 2;
    for (int q = tid; q < quadsA; q += 256) {
        int rr = q / kq;
        int kk = (q - rr * kq) * 4;
        int srow = row0 + rr; if (srow >= N) srow = N - 1;
        const float* gp = &A[(size_t)srow * K + kk];
        // low 32 bits of a generic pointer to LDS == byte offset in LDS
        unsigned ldsOff = (unsigned)(uintptr_t)&As[rr * strideA + kk];
        asm volatile("global_load_async_to_lds_b128 %0, %1, off"
                     :: "v"(ldsOff), "v"(gp) : "memory");
    }
    // ---- repack W slice through VGPRs (overlaps with async A transfers)
    for (int q = tid; q < K * 16; q += 256) {
        int k  = q >> 4;
        int c4 = (q & 15) * 4;
        float4 w = *(const float4*)&W[(size_t)k * HDIM + colBase + c4];
        int base = (k >> 1) * 128 + (k & 1);
        Wp[base + (c4 + 0) * 2] = w.x;
        Wp[base + (c4 + 1) * 2] = w.y;
        Wp[base + (c4 + 2) * 2] = w.z;
        Wp[base + (c4 + 3) * 2] = w.w;
    }
    asm volatile("s_wait_asynccnt 0x0" ::: "memory");   // A DMAs landed in LDS
    __syncthreads();

    const int aRow = rt * 16 + r;
    const int bCol = ct * 16 + r;

    v8f c = {};
    #pragma unroll 4
    for (int k0 = 0; k0 < K; k0 += 4) {
        const int ka = k0 + 2 * kh;
        v2f a = *(const v2f*)&As[aRow * strideA + ka];
        v2f b = *(const v2f*)&Wp[(ka >> 1) * 128 + bCol * 2];
        c = __builtin_amdgcn_wmma_f32_16x16x4_f32(
                /*neg_a=*/false, a, /*neg_b=*/false, b,
                /*c_mod=*/(short)0, c, /*reuse_a=*/false, /*reuse_b=*/false);
    }

    const int col = colBase + ct * 16 + r;
    #pragma unroll
    for (int j = 0; j < 8; ++j) {
        const int orow = row0 + rt * 16 + j + 8 * kh;
        if (orow < N) H[(size_t)orow * HDIM + col] = c[j];
    }
}

// ---------------------------------------------------------------- GCN aggregation
// out[i][c] = b[c] + h[i][c]*dinv[i]^2   (self-loop term + bias)
__global__ void agg_init(const float* __restrict__ h, const float* __restrict__ dinv,
                         const float* __restrict__ bias, float* __restrict__ out, int N) {
    int i = blockIdx.x * blockDim.x + threadIdx.x;
    if (i < N * HDIM) {
        int row = i >> 7, c = i & (HDIM - 1);
        float di = dinv[row];
        out[i] = bias[c] + h[i] * di * di;
    }
}

// out[d] += h[s] * dinv[s]*dinv[d]; one wave per edge, float4 per lane.
__global__ void agg_scatter(const float* __restrict__ h, const int* __restrict__ src,
                            const int* __restrict__ dst, const float* __restrict__ dinv,
                            float* __restrict__ out, int E) {
    int e  = blockIdx.x * 8 + (threadIdx.x >> 5);
    int c4 = (threadIdx.x & 31) * 4;
    if (e < E) {
        int s = src[e], d = dst[e];
        if ((threadIdx.x & 31) == 0 && e + 8 < E)   // warm next edge's source row
            __builtin_prefetch(&h[(size_t)src[e + 8] * HDIM], 0, 0);
        float coef = dinv[s] * dinv[d];
        float4 hv = *(const float4*)&h[(size_t)s * HDIM + c4];
        float* op = &out[(size_t)d * HDIM + c4];
        atomicAdd(op + 0, hv.x * coef);
        atomicAdd(op + 1, hv.y * coef);
        atomicAdd(op + 2, hv.z * coef);
        atomicAdd(op + 3, hv.w * coef);
    }
}

// ---------------------------------------------------------------- batch norm
// block handles 128 rows; thread (sub, c) accumulates rows sub, sub+2, ...
__global__ void bn_stats(const float* __restrict__ x, float* __restrict__ sum,
                         float* __restrict__ sq, int N) {
    int c   = threadIdx.x & (HDIM - 1);
    int sub = threadIdx.x >> 7;
    int rend = min((blockIdx.x + 1) * 128, N);
    float s = 0.0f, q = 0.0f;
    for (int r = blockIdx.x * 128 + sub; r < rend; r += 2) {
        float v = x[(size_t)r * HDIM + c];
        s += v; q += v * v;
    }
    atomicAdd(&sum[c], s);
    atomicAdd(&sq[c], q);
}

__global__ void bn_relu_apply(float* __restrict__ x, const float* __restrict__ sum,
                              const float* __restrict__ sq, const float* __restrict__ gam,
                              const float* __restrict__ bet, int N) {
    int i = blockIdx.x * blockDim.x + threadIdx.x;
    if (i < N * HDIM) {
        int c = i & (HDIM - 1);
        float invn = 1.0f / (float)N;
        float m = sum[c] * invn;
        float v = sq[c] * invn - m * m;
        float y = (x[i] - m) * rsqrtf(v + BN_EPS) * gam[c] + bet[c];
        x[i] = fmaxf(y, 0.0f);
    }
}

// ---------------------------------------------------------------- pooling
__global__ void pool_count(const int* __restrict__ batch, float* __restrict__ cnt, int N) {
    int i = blockIdx.x * blockDim.x + threadIdx.x;
    if (i < N) atomicAdd(&cnt[batch[i]], 1.0f);
}

// values are post-ReLU (>= 0) so int-bit atomicMax == float max; buffers zero-init.
__global__ void pool_scatter(const float* __restrict__ act, const int* __restrict__ batch,
                             float* __restrict__ psum, float* __restrict__ pmax, int N) {
    int i = blockIdx.x * blockDim.x + threadIdx.x;
    if (i < N * HDIM) {
        int row = i >> 7, c = i & (HDIM - 1);
        int g = batch[row];
        float v = act[i];
        atomicAdd(&psum[g * HDIM + c], v);
        atomicMax((int*)&pmax[g * HDIM + c], __float_as_int(v));
    }
}

// ---------------------------------------------------------------- MLP heads
__global__ void head_shared(const float* __restrict__ psum, const float* __restrict__ pmax,
                            const float* __restrict__ cnt, const float* __restrict__ Ws,
                            const float* __restrict__ bs, float* __restrict__ sh) {
    int i = blockIdx.x * blockDim.x + threadIdx.x;   // NGRAPHS*128
    if (i >= NGRAPHS * HDIM) return;
    int g = i >> 7, c = i & (HDIM - 1);
    float inv = 1.0f / fmaxf(cnt[g], 1.0f);
    float acc = bs[c];
    for (int k = 0; k < HDIM; ++k) acc += psum[g * HDIM + k] * inv * Ws[k * HDIM + c];
    for (int k = 0; k < HDIM; ++k) acc += pmax[g * HDIM + k] * Ws[(HDIM + k) * HDIM + c];
    sh[i] = fmaxf(acc, 0.0f);
}

__global__ void head_mid(const float* __restrict__ sh, const float* __restrict__ W,
                         const float* __restrict__ b, float* __restrict__ out) {
    int i = blockIdx.x * blockDim.x + threadIdx.x;   // NGRAPHS*32
    if (i >= NGRAPHS * 32) return;
    int g = i >> 5, j = i & 31;
    float acc = b[j];
    for (int k = 0; k < HDIM; ++k) acc += sh[g * HDIM + k] * W[k * 32 + j];
    out[i] = fmaxf(acc, 0.0f);
}

__global__ void head_final(const float* __restrict__ hd, const float* __restrict__ ht,
                           const float* __restrict__ Wd2, const float* __restrict__ bd2,
                           const float* __restrict__ Wt2, const float* __restrict__ bt2,
                           float* __restrict__ out) {
    int g = blockIdx.x * blockDim.x + threadIdx.x;
    if (g >= NGRAPHS) return;
    float a = bd2[0], b = bt2[0];
    for (int k = 0; k < 32; ++k) {
        a += hd[g * 32 + k] * Wd2[k];
        b += ht[g * 32 + k] * Wt2[k];
    }
    out[g] = a;
    out[NGRAPHS + g] = b;
}

// ---------------------------------------------------------------- launch
extern "C" void kernel_launch(void* const* d_in, const int* in_sizes, int n_in,
                              void* d_out, int out_size, void* d_ws, size_t ws_size,
                              hipStream_t stream) {
    const float* x     = (const float*)d_in[0];
    const int*   eidx  = (const int*)d_in[1];
    const int*   batch = (const int*)d_in[2];
    const float* Wl[4] = { (const float*)d_in[3], (const float*)d_in[5],
                           (const float*)d_in[7], (const float*)d_in[9] };
    const float* bl[4] = { (const float*)d_in[4], (const float*)d_in[6],
                           (const float*)d_in[8], (const float*)d_in[10] };
    const float* gam[4] = { (const float*)d_in[11], (const float*)d_in[13],
                            (const float*)d_in[15], (const float*)d_in[17] };
    const float* bet[4] = { (const float*)d_in[12], (const float*)d_in[14],
                            (const float*)d_in[16], (const float*)d_in[18] };
    const float* Ws  = (const float*)d_in[19];
    const float* bs  = (const float*)d_in[20];
    const float* Wd1 = (const float*)d_in[21];
    const float* bd1 = (const float*)d_in[22];
    const float* Wd2 = (const float*)d_in[23];
    const float* bd2 = (const float*)d_in[24];
    const float* Wt1 = (const float*)d_in[25];
    const float* bt1 = (const float*)d_in[26];
    const float* Wt2 = (const float*)d_in[27];
    const float* bt2 = (const float*)d_in[28];

    const int N = in_sizes[0] / 8;          // 50000
    const int E = in_sizes[1] / 2;          // 800000
    const int* src = eidx;
    const int* dst = eidx + E;

    // workspace layout (f32)
    float* bufA  = (float*)d_ws;            // N*128 : GEMM output h
    float* bufB  = bufA + (size_t)N * HDIM; // N*128 : aggregation / activation
    float* dinv  = bufB + (size_t)N * HDIM; // N
    float* chsum = dinv + N;                // 128
    float* chsq  = chsum + HDIM;            // 128
    float* psum  = chsq + HDIM;             // G*128
    float* pmax  = psum + NGRAPHS * HDIM;   // G*128
    float* cnt   = pmax + NGRAPHS * HDIM;   // G
    float* shbuf = cnt + NGRAPHS;           // G*128
    float* hd    = shbuf + NGRAPHS * HDIM;  // G*32
    float* ht    = hd + NGRAPHS * 32;       // G*32

    const int T = 256;
    const int NH = N * HDIM;

    // degrees -> dinv  (self-loop contributes 1)
    fill_f32<<<(N + T - 1) / T, T, 0, stream>>>(dinv, 1.0f, N);
    deg_scatter<<<(E + T - 1) / T, T, 0, stream>>>(dst, dinv, E);
    deg_finalize<<<(N + T - 1) / T, T, 0, stream>>>(dinv, N);

    // four GCN layers
    dim3 ggrid((N + 31) / 32, 2);
    for (int l = 0; l < 4; ++l) {
        const float* act = (l == 0) ? x : bufB;
        const int K = (l == 0) ? 8 : HDIM;
        gemm_wmma_f32<<<ggrid, 256, 0, stream>>>(act, Wl[l], bufA, N, K);
        agg_init<<<(NH + T - 1) / T, T, 0, stream>>>(bufA, dinv, bl[l], bufB, N);
        agg_scatter<<<(E + 7) / 8, 256, 0, stream>>>(bufA, src, dst, dinv, bufB, E);
        fill_f32<<<1, 256, 0, stream>>>(chsum, 0.0f, 2 * HDIM);   // chsum+chsq contiguous
        bn_stats<<<(N + 127) / 128, 256, 0, stream>>>(bufB, chsum, chsq, N);
        bn_relu_apply<<<(NH + T - 1) / T, T, 0, stream>>>(bufB, chsum, chsq, gam[l], bet[l], N);
    }

    // pooling (psum, pmax, cnt contiguous -> one zero fill; 0.0f bits == 0 for pmax init)
    fill_f32<<<(2 * NGRAPHS * HDIM + NGRAPHS + T - 1) / T, T, 0, stream>>>(
        psum, 0.0f, 2 * NGRAPHS * HDIM + NGRAPHS);
    pool_count<<<(N + T - 1) / T, T, 0, stream>>>(batch, cnt, N);
    pool_scatter<<<(NH + T - 1) / T, T, 0, stream>>>(bufB, batch, psum, pmax, N);

    // heads
    head_shared<<<(NGRAPHS * HDIM + T - 1) / T, T, 0, stream>>>(psum, pmax, cnt, Ws, bs, shbuf);
    head_mid<<<(NGRAPHS * 32 + T - 1) / T, T, 0, stream>>>(shbuf, Wd1, bd1, hd);
    head_mid<<<(NGRAPHS * 32 + T - 1) / T, T, 0, stream>>>(shbuf, Wt1, bt1, ht);
    head_final<<<1, 64, 0, stream>>>(hd, ht, Wd2, bd2, Wt2, bt2, (float*)d_out);
}